// GraphConvolutionBase_292057776282
// MI455X (gfx1250) — compile-verified
//
#include <hip/hip_runtime.h>

typedef float v2f __attribute__((ext_vector_type(2)));
typedef float v8f __attribute__((ext_vector_type(8)));

#define FEAT 256

// ---------------- deg[col] += 1 per edge ----------------
__global__ void deg_kernel(const int* __restrict__ col, float* __restrict__ deg, int E) {
    int e = blockIdx.x * blockDim.x + threadIdx.x;
    if (e < E) atomicAdd(&deg[col[e]], 1.0f);
}

// ---------------- Wsum = W[256:512] + Wr[256:512] ----------------
__global__ void wsum_kernel(const float* __restrict__ w, const float* __restrict__ wr,
                            float* __restrict__ wsum) {
    int i = blockIdx.x * blockDim.x + threadIdx.x;   // 0 .. 65535
    wsum[i] = w[FEAT * FEAT + i] + wr[FEAT * FEAT + i];
}

// ------- hi[col] += rsqrt(deg[col])*rsqrt(deg[row]) * x[row]  (4 edges/block) -------
__global__ void agg_kernel(const int* __restrict__ row, const int* __restrict__ col,
                           const float* __restrict__ x, const float* __restrict__ deg,
                           float* __restrict__ hi, int E) {
    int tid = threadIdx.x;
    int e = blockIdx.x * 4 + (tid >> 6);
    if (e >= E) return;
    int r = row[e];
    int c = col[e];
    float dc = deg[c];          // d_in
    float dr = deg[r];          // d_out
    if (!(dc > 0.0f) || !(dr > 0.0f)) return;
    float coef = rsqrtf(dc) * rsqrtf(dr);
    int f4 = (tid & 63) * 4;
    const float4 v = *reinterpret_cast<const float4*>(x + (size_t)r * FEAT + f4);
    float* dst = hi + (size_t)c * FEAT + f4;
    atomicAdd(dst + 0, coef * v.x);
    atomicAdd(dst + 1, coef * v.y);
    atomicAdd(dst + 2, coef * v.z);
    atomicAdd(dst + 3, coef * v.w);
}

// ------- out[m0:m0+16, 0:256] = hi@W0 + x@Wr0 + x0@Wsum  via V_WMMA_F32_16X16X4_F32 -------
// One wave per 16-row strip; 16 v8f accumulators cover all 256 output columns.
__global__ void __launch_bounds__(32) gemm_kernel(const float* __restrict__ hi,
                                                  const float* __restrict__ x,
                                                  const float* __restrict__ x0,
                                                  const float* __restrict__ w0,
                                                  const float* __restrict__ wr0,
                                                  const float* __restrict__ wsum,
                                                  float* __restrict__ out) {
    const int lane = threadIdx.x;            // wave32
    const int n    = lane & 15;              // column within 16-wide N tile
    const int hi16 = lane >> 4;              // 0 for lanes 0-15, 1 for lanes 16-31
    const int koff = hi16 << 1;              // A/B: lanes 16-31 carry K+2,K+3
    const int m0   = blockIdx.x * 16;

    v8f acc[16];
#pragma unroll
    for (int i = 0; i < 16; ++i) acc[i] = (v8f)0.0f;

    const float* As[3] = {hi, x, x0};
    const float* Bs[3] = {w0, wr0, wsum};

#pragma unroll
    for (int t = 0; t < 3; ++t) {
        const float* __restrict__ arow = As[t] + (size_t)(m0 + n) * FEAT;
        const float* __restrict__ B    = Bs[t];
        for (int k0 = 0; k0 < FEAT; k0 += 4) {
            // A 16x4 tile: lane holds A[M=n][k0+koff], A[M=n][k0+koff+1]
            v2f a = *reinterpret_cast<const v2f*>(arow + k0 + koff);
            const float* __restrict__ brow = B + (size_t)(k0 + koff) * FEAT + n;
#pragma unroll
            for (int nt = 0; nt < 16; ++nt) {
                // B 4x16 tile (mirrors C/D striping): lane holds B[k0+koff][n], B[k0+koff+1][n]
                v2f b;
                b.x = brow[nt * 16];
                b.y = brow[FEAT + nt * 16];
                acc[nt] = __builtin_amdgcn_wmma_f32_16x16x4_f32(
                    /*neg_a=*/false, a, /*neg_b=*/false, b,
                    /*c_mod=*/(short)0, acc[nt],
                    /*reuse_a=*/false, /*reuse_b=*/false);
            }
        }
    }

    // C/D layout: VGPR v -> M = v + (lane<16 ? 0 : 8), N = lane%16
    const int mbase = m0 + (hi16 << 3);
#pragma unroll
    for (int v = 0; v < 8; ++v) {
        float* __restrict__ orow = out + (size_t)(mbase + v) * FEAT + n;
#pragma unroll
        for (int nt = 0; nt < 16; ++nt) {
            orow[nt * 16] = acc[nt][v];
        }
    }
}

extern "C" void kernel_launch(void* const* d_in, const int* in_sizes, int n_in,
                              void* d_out, int out_size, void* d_ws, size_t ws_size,
                              hipStream_t stream) {
    const float* x  = (const float*)d_in[0];
    const int*   ei = (const int*)d_in[1];
    const float* x0 = (const float*)d_in[2];
    const float* w  = (const float*)d_in[3];
    const float* wr = (const float*)d_in[4];
    float* out = (float*)d_out;

    const int N = in_sizes[0] / FEAT;   // 100000
    const int E = in_sizes[1] / 2;      // 1600000
    const int* row = ei;
    const int* col = ei + E;

    // workspace layout: hi[N*256] | deg[N] | wsum[256*256]
    float* hi   = (float*)d_ws;
    float* deg  = hi + (size_t)N * FEAT;
    float* wsum = deg + N;

    // zero hi + deg (wsum fully overwritten)
    hipMemsetAsync(d_ws, 0, ((size_t)N * FEAT + (size_t)N) * sizeof(float), stream);

    wsum_kernel<<<FEAT, FEAT, 0, stream>>>(w, wr, wsum);
    deg_kernel<<<(E + 255) / 256, 256, 0, stream>>>(col, deg, E);
    agg_kernel<<<(E + 3) / 4, 256, 0, stream>>>(row, col, x, deg, hi, E);
    gemm_kernel<<<N / 16, 32, 0, stream>>>(hi, x, x0, w, wr, wsum, out);
}